// EfficientDet_45509473468576
// MI455X (gfx1250) — compile-verified
//
#include <hip/hip_runtime.h>
#include <cfloat>

#define BB 16
#define AA 49104
#define CC 90
#define PRE_K 1000
#define MAX_DET 100
#define CONF 0.05f
#define IOU_TH 0.5f
#define IMG 512.0f
#define CLS_OFF 4096.0f

#define T1 128   // phase-1 block size / anchors per tile
#define T2 1024  // phase-2 block size
#define CAP 2048 // candidate gather capacity

// ---------------------------------------------------------------------------
// Phase 1: stream classification tiles (contiguous 45KB chunks) into LDS via
// CDNA5 async global->LDS DMA, reduce max/argmax over 90 classes per anchor.
// Bandwidth-bound: 283MB read -> ~12us floor at 23.3 TB/s.
// ---------------------------------------------------------------------------
__global__ __launch_bounds__(T1) void score_reduce_kernel(
    const float* __restrict__ cls, float* __restrict__ wsScore,
    int* __restrict__ wsCls)
{
    __shared__ alignas(16) float tile[T1 * CC];  // 45 KB
    const int b   = blockIdx.y;
    const int a0  = blockIdx.x * T1;
    const int na  = min(T1, AA - a0);
    const int tid = threadIdx.x;

    // classification[b, a0 .. a0+na, :] is one fully contiguous, 16B-aligned
    // region: DMA it into LDS with async b128 transfers (ASYNCcnt tracked).
    const unsigned long long gsrc =
        (unsigned long long)(const void*)(cls + ((size_t)b * AA + a0) * CC);
    const unsigned lbase = (unsigned)(unsigned long long)(const void*)&tile[0];
    const int nbytes = na * CC * 4;  // multiple of 16 (na even, 360B rows)
    for (int off = tid * 16; off < nbytes; off += T1 * 16) {
        unsigned long long ga = gsrc + (unsigned)off;
        unsigned la = lbase + (unsigned)off;
        asm volatile("global_load_async_to_lds_b128 %0, %1, off"
                     :: "v"(la), "v"(ga) : "memory");
    }
    asm volatile("s_wait_asynccnt 0" ::: "memory");
    __syncthreads();

    if (tid < na) {
        // Row stride = 90 dwords -> lanes hit 32 distinct LDS banks (gcd(90,64)=2).
        const float2* row =
            reinterpret_cast<const float2*>(&tile[(size_t)tid * CC]);
        float best = -FLT_MAX;
        int bi = 0;
        #pragma unroll 5
        for (int k = 0; k < CC / 2; ++k) {
            float2 v = row[k];
            if (v.x > best) { best = v.x; bi = 2 * k; }        // first-max
            if (v.y > best) { best = v.y; bi = 2 * k + 1; }    // (jnp.argmax)
        }
        size_t o = (size_t)b * AA + a0 + tid;
        wsScore[o] = (best > CONF) ? best : -1.0f;  // reference's pre-top_k mask
        wsCls[o]   = bi;
    }
}

// monotone key: larger key <=> larger float (handles the -1 sentinel too)
__device__ __forceinline__ unsigned fkey(float f) {
    unsigned u = __float_as_uint(f);
    return (u & 0x80000000u) ? ~u : (u | 0x80000000u);
}

// ---------------------------------------------------------------------------
// Phase 2: one 1024-thread workgroup per image.
//  - radix-select the rank-1000 score threshold (4x 256-bin passes, L2 data)
//  - gather >= threshold, bitonic-sort 2048 by (score desc, idx asc)
//  - async-gather anchors+regression (16B/lane scatter-gather) and decode
//  - greedy class-offset NMS (order-dependent, 1000 barrier iterations)
//  - prefix-sum compaction -> top-100 output with pad/scale/clip
// ---------------------------------------------------------------------------
__global__ __launch_bounds__(T2) void select_nms_kernel(
    const float* __restrict__ anchors, const float* __restrict__ reg,
    const float* __restrict__ scales, const float* __restrict__ pads,
    const float* __restrict__ wsScore, const int* __restrict__ wsCls,
    float* __restrict__ out)
{
    __shared__ unsigned sh_hist[256];
    __shared__ unsigned sh_prefix;
    __shared__ unsigned sh_remain;
    __shared__ int sh_cnt;
    __shared__ alignas(16) float skey[CAP];
    __shared__ int sidx[CAP];
    __shared__ int ccls[T2];
    __shared__ int keepf[T2];
    __shared__ alignas(16) union Pool {
        struct { float ganc[T2 * 4]; float greg[T2 * 4]; } g;  // 32 KB gather
        struct { float x1[T2]; float y1[T2]; float x2[T2]; float y2[T2]; } bx;
    } pool;

    const int b   = blockIdx.x;
    const int tid = threadIdx.x;
    const float* sc = wsScore + (size_t)b * AA;

    if (tid == 0) { sh_prefix = 0u; sh_remain = PRE_K; sh_cnt = 0; }

    // ---- radix select: find key of the 1000th-largest score ----
    for (int pass = 0; pass < 4; ++pass) {
        const int shift = 24 - 8 * pass;
        for (int i = tid; i < 256; i += T2) sh_hist[i] = 0u;
        __syncthreads();
        const unsigned prefix = sh_prefix;
        const unsigned mask = (pass == 0) ? 0u : (0xFFFFFFFFu << (shift + 8));
        for (int a = tid; a < AA; a += T2) {
            unsigned u = fkey(sc[a]);
            if ((u & mask) == prefix)
                atomicAdd(&sh_hist[(u >> shift) & 0xFFu], 1u);
        }
        __syncthreads();
        if (tid == 0) {
            unsigned rem = sh_remain, cum = 0;
            int d = 255;
            while (d > 0) {
                unsigned c = sh_hist[d];
                if (cum + c >= rem) break;
                cum += c; --d;
            }
            sh_remain = rem - cum;
            sh_prefix = prefix | ((unsigned)d << shift);
        }
        __syncthreads();
    }

    // ---- gather all candidates with key >= threshold ----
    const unsigned T = sh_prefix;
    for (int a = tid; a < AA; a += T2) {
        float s = sc[a];
        if (fkey(s) >= T) {
            int p = atomicAdd(&sh_cnt, 1);
            if (p < CAP) { skey[p] = s; sidx[p] = a; }
        }
    }
    __syncthreads();
    const int cnt = min(sh_cnt, CAP);
    for (int i = tid; i < CAP; i += T2)
        if (i >= cnt) { skey[i] = -FLT_MAX; sidx[i] = 0x7FFFFFFF; }

    // ---- bitonic sort 2048 by (score desc, index asc) == lax.top_k order ----
    for (int k = 2; k <= CAP; k <<= 1) {
        for (int j = k >> 1; j > 0; j >>= 1) {
            __syncthreads();
            for (int i = tid; i < CAP; i += T2) {
                int ixj = i ^ j;
                if (ixj > i) {
                    float s1 = skey[i], s2 = skey[ixj];
                    int d1 = sidx[i], d2 = sidx[ixj];
                    bool iWorse = (s2 > s1) || ((s2 == s1) && (d2 < d1));
                    bool dirAsc = ((i & k) == 0);
                    if (iWorse == dirAsc) {
                        skey[i] = s2; sidx[i] = d2;
                        skey[ixj] = s1; sidx[ixj] = d1;
                    }
                }
            }
        }
    }
    __syncthreads();

    // ---- async scatter-gather: 16B anchor + regression records per lane ----
    const int t = tid;
    if (t < PRE_K) {
        int a = sidx[t];
        unsigned la1 = (unsigned)(unsigned long long)(const void*)&pool.g.ganc[t * 4];
        unsigned la2 = (unsigned)(unsigned long long)(const void*)&pool.g.greg[t * 4];
        unsigned long long ga1 =
            (unsigned long long)(const void*)(anchors + (size_t)a * 4);
        unsigned long long ga2 =
            (unsigned long long)(const void*)(reg + ((size_t)b * AA + a) * 4);
        asm volatile("global_load_async_to_lds_b128 %0, %1, off"
                     :: "v"(la1), "v"(ga1) : "memory");
        asm volatile("global_load_async_to_lds_b128 %0, %1, off"
                     :: "v"(la2), "v"(ga2) : "memory");
        ccls[t] = wsCls[(size_t)b * AA + a];
    } else {
        ccls[t] = -1;
    }
    asm volatile("s_wait_asynccnt 0" ::: "memory");
    __syncthreads();

    float4 an = make_float4(0.f, 0.f, 0.f, 0.f);
    float4 rg = make_float4(0.f, 0.f, 0.f, 0.f);
    if (t < PRE_K) {
        an = *reinterpret_cast<const float4*>(&pool.g.ganc[t * 4]);
        rg = *reinterpret_cast<const float4*>(&pool.g.greg[t * 4]);
    }
    __syncthreads();  // pool union: all reads of .g done before writes to .bx

    if (t < PRE_K) {
        // decode (anchors are y1,x1,y2,x2; regression dy,dx,dh,dw)
        float yca = (an.x + an.z) * 0.5f;
        float xca = (an.y + an.w) * 0.5f;
        float ha = an.z - an.x, wa = an.w - an.y;
        float w = expf(rg.w) * wa;
        float h = expf(rg.z) * ha;
        float yc = rg.x * ha + yca;
        float xc = rg.y * wa + xca;
        pool.bx.x1[t] = xc - w * 0.5f;
        pool.bx.y1[t] = yc - h * 0.5f;
        pool.bx.x2[t] = xc + w * 0.5f;
        pool.bx.y2[t] = yc + h * 0.5f;
        keepf[t] = (skey[t] > CONF) ? 1 : 0;
    } else {
        pool.bx.x1[t] = 0.f; pool.bx.y1[t] = 0.f;
        pool.bx.x2[t] = 0.f; pool.bx.y2[t] = 0.f;
        keepf[t] = 0;
    }
    __syncthreads();

    // ---- greedy NMS: class-offset boxes, score-sorted order ----
    for (int i = 0; i < PRE_K; ++i) {
        __syncthreads();
        if (keepf[i] == 0) continue;  // uniform (LDS broadcast)
        float offi = (float)ccls[i] * CLS_OFF;
        float xi1 = pool.bx.x1[i] + offi, yi1 = pool.bx.y1[i] + offi;
        float xi2 = pool.bx.x2[i] + offi, yi2 = pool.bx.y2[i] + offi;
        float ai = fmaxf(xi2 - xi1, 0.f) * fmaxf(yi2 - yi1, 0.f);
        if (t > i && t < PRE_K && keepf[t]) {
            float offj = (float)ccls[t] * CLS_OFF;
            float xj1 = pool.bx.x1[t] + offj, yj1 = pool.bx.y1[t] + offj;
            float xj2 = pool.bx.x2[t] + offj, yj2 = pool.bx.y2[t] + offj;
            float aj = fmaxf(xj2 - xj1, 0.f) * fmaxf(yj2 - yj1, 0.f);
            float lx = fmaxf(xi1, xj1), ly = fmaxf(yi1, yj1);
            float rx = fminf(xi2, xj2), ry = fminf(yi2, yj2);
            float inter = fmaxf(rx - lx, 0.f) * fmaxf(ry - ly, 0.f);
            float iou = inter / fmaxf(ai + aj - inter, 1e-8f);
            if (iou > IOU_TH) keepf[t] = 0;
        }
    }
    __syncthreads();

    // ---- stable compaction of kept candidates (Hillis-Steele scan) ----
    sidx[t] = keepf[t];
    __syncthreads();
    for (int off = 1; off < T2; off <<= 1) {
        int v = sidx[t];
        if (t >= off) v += sidx[t - off];
        __syncthreads();
        sidx[t] = v;
        __syncthreads();
    }
    const int incl = sidx[t];
    const int excl = incl - keepf[t];
    const int nk = min(sidx[T2 - 1], MAX_DET);

    // ---- write outputs ----
    const float sy = scales[b * 2 + 0], sx = scales[b * 2 + 1];
    const float py = pads[b * 2 + 0],  px = pads[b * 2 + 1];
    float* outLab = out + (size_t)BB * MAX_DET * 5;  // labels follow out5 flat

    if (keepf[t] && excl < MAX_DET) {
        float x1 = fminf(fmaxf((pool.bx.x1[t] - px) / sx, 0.f), IMG);
        float y1 = fminf(fmaxf((pool.bx.y1[t] - py) / sy, 0.f), IMG);
        float x2 = fminf(fmaxf((pool.bx.x2[t] - px) / sx, 0.f), IMG);
        float y2 = fminf(fmaxf((pool.bx.y2[t] - py) / sy, 0.f), IMG);
        float* o = out + ((size_t)b * MAX_DET + excl) * 5;
        o[0] = x1; o[1] = y1; o[2] = x2; o[3] = y2; o[4] = skey[t];
        outLab[(size_t)b * MAX_DET + excl] = (float)ccls[t];
    }
    if (t >= nk && t < MAX_DET) {  // invalid slots -> zeros, label -1
        float* o = out + ((size_t)b * MAX_DET + t) * 5;
        o[0] = 0.f; o[1] = 0.f; o[2] = 0.f; o[3] = 0.f; o[4] = 0.f;
        outLab[(size_t)b * MAX_DET + t] = -1.0f;
    }
}

extern "C" void kernel_launch(void* const* d_in, const int* in_sizes, int n_in,
                              void* d_out, int out_size, void* d_ws, size_t ws_size,
                              hipStream_t stream) {
    const float* anchors        = (const float*)d_in[0];
    const float* regression     = (const float*)d_in[1];
    const float* classification = (const float*)d_in[2];
    const float* scales         = (const float*)d_in[3];
    const float* pads           = (const float*)d_in[4];
    float* out = (float*)d_out;

    // scratch: masked max-scores [B,A] f32 + argmax class [B,A] i32 (~6.3 MB)
    float* wsScore = (float*)d_ws;
    int* wsCls = (int*)((char*)d_ws + (size_t)BB * AA * sizeof(float));

    dim3 g1((AA + T1 - 1) / T1, BB);
    score_reduce_kernel<<<g1, T1, 0, stream>>>(classification, wsScore, wsCls);
    select_nms_kernel<<<BB, T2, 0, stream>>>(anchors, regression, scales, pads,
                                             wsScore, wsCls, out);
    (void)in_sizes; (void)n_in; (void)out_size; (void)ws_size;
}